// GCNSampling_67542655697275
// MI455X (gfx1250) — compile-verified
//
#include <hip/hip_runtime.h>

// GCN 2-layer: segment-mean -> GEMM(+bias,ReLU) -> segment-mean -> GEMM(+bias)
// N=100000 nodes, D=128 feats, H=128 hidden, C=64 classes, E=1.6M edges.
// Bottleneck: edge aggregation (float4 gather + L2-resident f32 atomics).
// GEMMs: fp32 WMMA (V_WMMA_F32_16X16X4_F32); one wave per 16-row m-tile
// computes all n-tiles (A read exactly once, 8 WMMAs per A load). W is staged
// in LDS k-pair interleaved (+XOR-16 column swizzle) so each B operand is one
// bank-conflict-free ds_load_b64; segment-mean divide fused into the A load;
// A loads software-pipelined one k-step ahead.

#define NNODES 100000
#define DFEAT  128

typedef __attribute__((ext_vector_type(2))) float v2f;
typedef __attribute__((ext_vector_type(8))) float v8f;

// ---------------------------------------------------------------- utilities
__global__ void fill_zero_kernel(float* __restrict__ p, long long n) {
  long long i = (long long)blockIdx.x * blockDim.x + threadIdx.x;
  long long stride = (long long)gridDim.x * blockDim.x;
  for (; i < n; i += stride) p[i] = 0.0f;
}

__global__ void count_kernel(const int* __restrict__ dst,
                             float* __restrict__ cnt, int E) {
  int i = blockIdx.x * blockDim.x + threadIdx.x;
  if (i < E) unsafeAtomicAdd(&cnt[dst[i]], 1.0f);   // global_atomic_add_f32
}

// One wave (32 lanes) per edge; each lane moves 16B (float4) of the 512B row.
__global__ void scatter_add_kernel(const float* __restrict__ X,   // [N,128]
                                   const int* __restrict__ src,
                                   const int* __restrict__ dst,
                                   float* __restrict__ acc,       // [N,128]
                                   int E) {
  long long t = (long long)blockIdx.x * blockDim.x + threadIdx.x;
  long long e = t >> 5;
  if (e >= E) return;
  const int lane = (int)(t & 31);
  const int s = src[e];
  const int d = dst[e];
  const float4 v = ((const float4*)(X + (long long)s * DFEAT))[lane]; // b128
  float* a = acc + (long long)d * DFEAT + lane * 4;
  unsafeAtomicAdd(a + 0, v.x);
  unsafeAtomicAdd(a + 1, v.y);
  unsafeAtomicAdd(a + 2, v.z);
  unsafeAtomicAdd(a + 3, v.w);
}

// ---------------------------------------------------------------- fp32 WMMA
// D(16x16,f32) = A(16x4,f32) * B(4x16,f32) + C  — VOP3P, wave32.
__device__ __forceinline__ v8f wmma16x16x4(v2f a, v2f b, v8f c) {
  return __builtin_amdgcn_wmma_f32_16x16x4_f32(false, a, false, b,
                                               (short)0, c, false, false);
}

// out = (A / max(cnt,1)) @ W + bias, optional ReLU.
// Block = 8 waves; wave w owns m-tile (blockIdx*8 + w) and all NT n-tiles.
// LDS layout: element W[k][c] at sW[(k>>1)*2N + 2*csw + (k&1)], where
// csw = c ^ (((k>>1)&1)<<4). Since k steps by 4, rows {ka, ka+1} for the two
// lane halves share pair-parity == half, so {b.x, b.y} is one aligned v2f and
// the two halves read disjoint 32-bank groups.
template <int KDIM, int NDIM, bool RELU>
__global__ __launch_bounds__(256) void gemm_mean_bias_kernel(
    const float* __restrict__ A,     // [M,K] un-normalized segment sums
    const float* __restrict__ cnt,   // [M]
    const float* __restrict__ W,     // [K,N]
    const float* __restrict__ bias,  // [N]
    float* __restrict__ out,         // [M,N]
    int M) {
  constexpr int NT = NDIM / 16;
  constexpr int HALFK = KDIM / 2;
  __shared__ float sW[KDIM * NDIM];

  // Stage W: one thread handles one (k-pair, col) -> contiguous ds_store_b64.
  for (int i = threadIdx.x; i < HALFK * NDIM; i += blockDim.x) {
    const int p = i / NDIM;                 // k-pair index (rows 2p, 2p+1)
    const int c = i % NDIM;
    const int csw = c ^ ((p & 1) << 4);     // XOR-16 swizzle on odd pairs
    float2 v;
    v.x = W[(2 * p) * NDIM + c];
    v.y = W[(2 * p + 1) * NDIM + c];
    *(float2*)&sW[p * (2 * NDIM) + 2 * csw] = v;
  }
  __syncthreads();

  const int lane = threadIdx.x & 31;
  const int wave = threadIdx.x >> 5;
  const int m0 = (blockIdx.x * 8 + wave) << 4;
  if (m0 >= M) return;  // wave-uniform exit (after the barrier)

  const int half = lane >> 4;  // 0: lanes 0-15 (K=k,k+1), 1: lanes 16-31 (K=k+2,k+3)
  const int l = lane & 15;
  const float* arow = A + (long long)(m0 + l) * KDIM;
  const float rcp = 1.0f / fmaxf(cnt[m0 + l], 1.0f);   // fused segment-mean

  v8f acc[NT];
#pragma unroll
  for (int nt = 0; nt < NT; ++nt) acc[nt] = {};

  // software pipeline: prefetch A one k-step ahead
  v2f a_next = *(const v2f*)&arow[2 * half];            // ka for k=0
  for (int k = 0; k < KDIM; k += 4) {
    v2f a;
    a.x = a_next.x * rcp;
    a.y = a_next.y * rcp;
    if (k + 4 < KDIM)
      a_next = *(const v2f*)&arow[k + 4 + 2 * half];
    // row pair p = k/2 + half  (p & 1 == half because k % 4 == 0)
    const float* wbase = &sW[(k / 2 + half) * (2 * NDIM) + 2 * l];
#pragma unroll
    for (int nt = 0; nt < NT; ++nt) {
      const v2f b = *(const v2f*)&wbase[(nt ^ half) << 5]; // one ds_load_b64
      acc[nt] = wmma16x16x4(a, b, acc[nt]);
    }
  }

#pragma unroll
  for (int nt = 0; nt < NT; ++nt) {
    const float bv = bias[(nt << 4) + l];
#pragma unroll
    for (int r = 0; r < 8; ++r) {
      float v = acc[nt][r] + bv;
      if (RELU) v = fmaxf(v, 0.0f);
      out[(long long)(m0 + r + 8 * half) * NDIM + (nt << 4) + l] = v;
    }
  }
}

// ---------------------------------------------------------------- driver
extern "C" void kernel_launch(void* const* d_in, const int* in_sizes, int n_in,
                              void* d_out, int out_size, void* d_ws,
                              size_t ws_size, hipStream_t stream) {
  const float* features = (const float*)d_in[0];   // [100000,128]
  const float* W1       = (const float*)d_in[1];   // [128,128]
  const float* b1       = (const float*)d_in[2];   // [128]
  const float* W2       = (const float*)d_in[3];   // [128,64]
  const float* b2       = (const float*)d_in[4];   // [64]
  const int*   src      = (const int*)d_in[5];     // [E]
  const int*   dst      = (const int*)d_in[6];     // [E]
  const int E = in_sizes[5];

  // workspace: aggA [N*128] | h1 [N*128] | cnt [N]   (~103 MB)
  float* aggA = (float*)d_ws;
  float* h1   = aggA + (size_t)NNODES * DFEAT;
  float* cnt  = h1 + (size_t)NNODES * DFEAT;

  const long long ND = (long long)NNODES * DFEAT;
  const int scatter_blocks = (int)(((long long)E * 32 + 255) / 256);
  const int gemm_blocks = (NNODES / 16 + 7) / 8;   // 8 m-tiles per block

  // ---- layer 0: mean-aggregate features -> relu(h0 @ W1 + b1)
  fill_zero_kernel<<<4096, 256, 0, stream>>>(aggA, ND);
  fill_zero_kernel<<<512, 256, 0, stream>>>(cnt, NNODES);
  count_kernel<<<(E + 255) / 256, 256, 0, stream>>>(dst, cnt, E);
  scatter_add_kernel<<<scatter_blocks, 256, 0, stream>>>(features, src, dst,
                                                         aggA, E);
  gemm_mean_bias_kernel<128, 128, true>
      <<<gemm_blocks, 256, 0, stream>>>(aggA, cnt, W1, b1, h1, NNODES);

  // ---- layer 1: mean-aggregate h1 -> h @ W2 + b2
  fill_zero_kernel<<<4096, 256, 0, stream>>>(aggA, ND);
  scatter_add_kernel<<<scatter_blocks, 256, 0, stream>>>(h1, src, dst, aggA, E);
  gemm_mean_bias_kernel<128, 64, false>
      <<<gemm_blocks, 256, 0, stream>>>(aggA, cnt, W2, b2, (float*)d_out,
                                        NNODES);
}